// Wav2vec2ForSSL_3770981286079
// MI455X (gfx1250) — compile-verified
//
#include <hip/hip_runtime.h>
#include <math.h>

// Problem constants (wav2vec2 SSL loss)
#define B_ 8
#define T_ 1024
#define D_ 256
#define K_ 100
#define G_ 2
#define V_ 320

typedef __attribute__((ext_vector_type(2))) float v2f;
typedef __attribute__((ext_vector_type(8))) float v8f;

__device__ __forceinline__ float wave_sum(float v) {
#pragma unroll
  for (int off = 16; off > 0; off >>= 1) v += __shfl_xor(v, off, 32);
  return v;
}
__device__ __forceinline__ float wave_max(float v) {
#pragma unroll
  for (int off = 16; off > 0; off >>= 1) v = fmaxf(v, __shfl_xor(v, off, 32));
  return v;
}

// ---------------------------------------------------------------------------
// Zero the small accumulators: avg[G*V] followed by loss_c[1] (contiguous).
__global__ void init_kernel(float* acc) {
  if (threadIdx.x < G_ * V_ + 1) acc[threadIdx.x] = 0.0f;
}

// ---------------------------------------------------------------------------
// Per-row L2 norms of quantized and context features. One wave per row.
__global__ __launch_bounds__(256) void norms_kernel(const float* __restrict__ q,
                                                    const float* __restrict__ c,
                                                    float* __restrict__ qn,
                                                    float* __restrict__ cn) {
  const int wave = threadIdx.x >> 5, lane = threadIdx.x & 31;
  const int row = blockIdx.x * 8 + wave;  // 0 .. B*T-1
  const float* qp = q + (size_t)row * D_;
  const float* cp = c + (size_t)row * D_;
  float sq = 0.0f, sc = 0.0f;
#pragma unroll
  for (int i = lane; i < D_; i += 32) {
    float a = qp[i]; sq += a * a;
    float b = cp[i]; sc += b * b;
  }
  sq = wave_sum(sq);
  sc = wave_sum(sc);
  if (lane == 0) {
    qn[row] = sqrtf(sq);
    cn[row] = sqrtf(sc);
  }
}

// ---------------------------------------------------------------------------
// S[b] = C[b] * Q[b]^T : (T x D) x (D x T) -> (T x T), fp32 WMMA.
// Each wave owns a 32x32 output tile = 4 accumulators of V_WMMA_F32_16X16X4.
// 16x4 f32 A layout: lanes 0-15 hold {K=0,K=1}, lanes 16-31 hold {K=2,K=3};
// B (4x16) mirrors with lane = column. C/D: VGPR r holds M=r (lanes 0-15)
// and M=r+8 (lanes 16-31), N = lane&15.
// Batch selected by blockIdx.z via io_stride/s_stride (0 for per-batch path).
__global__ __launch_bounds__(256) void gemm_scores(const float* __restrict__ Cbase,
                                                   const float* __restrict__ Qbase,
                                                   float* __restrict__ Sbase,
                                                   long io_stride, long s_stride) {
  const float* Cm = Cbase + (size_t)blockIdx.z * io_stride;
  const float* Qm = Qbase + (size_t)blockIdx.z * io_stride;
  float* S = Sbase + (size_t)blockIdx.z * s_stride;

  const int wave = threadIdx.x >> 5;
  const int lane = threadIdx.x & 31;
  const int lr = lane & 15;
  const int hi = lane >> 4;  // 0: K=0,1  1: K=2,3
  const int m0 = blockIdx.y * 128 + (wave & 3) * 32;  // context rows (t)
  const int n0 = blockIdx.x * 64 + (wave >> 2) * 32;  // quantized rows (tau)

  const float* a0p = Cm + (size_t)(m0 + lr) * D_ + 2 * hi;
  const float* a1p = Cm + (size_t)(m0 + 16 + lr) * D_ + 2 * hi;
  const float* b0p = Qm + (size_t)(n0 + lr) * D_ + 2 * hi;
  const float* b1p = Qm + (size_t)(n0 + 16 + lr) * D_ + 2 * hi;

  v8f acc00 = {}, acc01 = {}, acc10 = {}, acc11 = {};

  // Software-pipelined: operands for step k+4 are in flight while the four
  // WMMAs for step k execute, keeping >=4 loads outstanding.
  v2f a0 = *(const v2f*)(a0p);
  v2f a1 = *(const v2f*)(a1p);
  v2f b0 = *(const v2f*)(b0p);
  v2f b1 = *(const v2f*)(b1p);
#pragma unroll 8
  for (int k = 0; k < D_; k += 4) {
    v2f na0 = a0, na1 = a1, nb0 = b0, nb1 = b1;
    if (k + 4 < D_) {
      na0 = *(const v2f*)(a0p + k + 4);
      na1 = *(const v2f*)(a1p + k + 4);
      nb0 = *(const v2f*)(b0p + k + 4);
      nb1 = *(const v2f*)(b1p + k + 4);
    }
    acc00 = __builtin_amdgcn_wmma_f32_16x16x4_f32(false, a0, false, b0, (short)0, acc00, false, false);
    acc01 = __builtin_amdgcn_wmma_f32_16x16x4_f32(false, a0, false, b1, (short)0, acc01, false, false);
    acc10 = __builtin_amdgcn_wmma_f32_16x16x4_f32(false, a1, false, b0, (short)0, acc10, false, false);
    acc11 = __builtin_amdgcn_wmma_f32_16x16x4_f32(false, a1, false, b1, (short)0, acc11, false, false);
    a0 = na0; a1 = na1; b0 = nb0; b1 = nb1;
  }

#pragma unroll
  for (int r = 0; r < 8; ++r) {
    const int row0 = m0 + r + 8 * hi;
    const int row1 = row0 + 16;
    const int col0 = n0 + lr;
    const int col1 = col0 + 16;
    S[(size_t)row0 * T_ + col0] = acc00[r];
    S[(size_t)row0 * T_ + col1] = acc01[r];
    S[(size_t)row1 * T_ + col0] = acc10[r];
    S[(size_t)row1 * T_ + col1] = acc11[r];
  }
}

// ---------------------------------------------------------------------------
// Contrastive log-softmax over (1 positive + K negatives) per (b,t).
// One wave per t; 101 logits covered by 4 strided chunks of 32 lanes.
// b = b_base + blockIdx.y ; scores for this batch at S + blockIdx.y*s_stride.
__global__ __launch_bounds__(256) void contrastive_kernel(
    const float* __restrict__ S, long s_stride, const int* __restrict__ neg_idx,
    const unsigned char* __restrict__ mask, const float* __restrict__ qn,
    const float* __restrict__ cn, float* __restrict__ loss_acc, int b_base) {
  const int wave = threadIdx.x >> 5, lane = threadIdx.x & 31;
  const int b = b_base + blockIdx.y;
  const int t = blockIdx.x * 8 + wave;
  const float* Srow = S + (size_t)blockIdx.y * s_stride + (size_t)t * T_;
  const int* idxp = neg_idx + ((size_t)(b * T_ + t)) * K_;
  const float pn = cn[b * T_ + t];

  float lg[4];
#pragma unroll
  for (int c = 0; c < 4; ++c) {
    const int j = lane + 32 * c;
    float v = -INFINITY;
    if (j <= K_) {
      float s, tn;
      bool coll = false;
      if (j == 0) {
        s = Srow[t];
        tn = qn[b * T_ + t];
      } else {
        const int idx = idxp[j - 1];
        s = Srow[idx];
        tn = qn[b * T_ + idx];
        coll = (idx == t);
      }
      const float cosv = s / fmaxf(pn * tn, 1e-8f);
      v = coll ? -INFINITY : cosv * 10.0f;  // divide by TEMP = 0.1
    }
    lg[c] = v;
  }
  float mx = fmaxf(fmaxf(lg[0], lg[1]), fmaxf(lg[2], lg[3]));
  mx = wave_max(mx);
  float se = 0.0f;
#pragma unroll
  for (int c = 0; c < 4; ++c) se += expf(lg[c] - mx);  // exp(-inf)=0
  se = wave_sum(se);
  const float logit0 = __shfl(lg[0], 0, 32);
  if (lane == 0 && mask[b * T_ + t]) {
    const float logp0 = logit0 - (mx + logf(se));
    atomicAdd(loss_acc, -logp0);
  }
}

// ---------------------------------------------------------------------------
// Codebook diversity: softmax over V per (b,t,g), accumulate probability sums
// into avg[G*V] (scaled by 1/(B*T) later). One wave per (row,g), LDS staging.
__global__ __launch_bounds__(256) void diversity_kernel(const float* __restrict__ logits,
                                                        float* __restrict__ avg_global) {
  __shared__ float lavg[G_ * V_];
  for (int i = threadIdx.x; i < G_ * V_; i += blockDim.x) lavg[i] = 0.0f;
  __syncthreads();

  const int wave = threadIdx.x >> 5, lane = threadIdx.x & 31;
  const int base = blockIdx.x * 64;  // 64 (b*T+t) rows per block
#pragma unroll
  for (int rr = 0; rr < 8; ++rr) {
    const int row = base + wave * 8 + rr;
#pragma unroll
    for (int g = 0; g < G_; ++g) {
      const float* lp = logits + ((size_t)row * G_ + g) * V_;
      float x[10];
      float mx = -INFINITY;
#pragma unroll
      for (int c = 0; c < 10; ++c) {
        x[c] = lp[lane + 32 * c];
        mx = fmaxf(mx, x[c]);
      }
      mx = wave_max(mx);
      float se = 0.0f;
#pragma unroll
      for (int c = 0; c < 10; ++c) {
        x[c] = expf(x[c] - mx);
        se += x[c];
      }
      se = wave_sum(se);
      const float inv = 1.0f / se;
#pragma unroll
      for (int c = 0; c < 10; ++c)
        atomicAdd(&lavg[g * V_ + lane + 32 * c], x[c] * inv);
    }
  }
  __syncthreads();
  for (int i = threadIdx.x; i < G_ * V_; i += blockDim.x)
    atomicAdd(&avg_global[i], lavg[i]);
}

// ---------------------------------------------------------------------------
// Final scalar: loss = loss_c + 0.1*loss_d + 10*loss_l2.
__global__ __launch_bounds__(256) void final_kernel(const float* __restrict__ qn,
                                                    const float* __restrict__ avg_raw,
                                                    const float* __restrict__ loss_c,
                                                    float* __restrict__ out) {
  __shared__ float ssum;
  __shared__ float ent[G_];
  if (threadIdx.x == 0) { ssum = 0.0f; ent[0] = 0.0f; ent[1] = 0.0f; }
  __syncthreads();

  float part = 0.0f;
  for (int i = threadIdx.x; i < B_ * T_; i += blockDim.x) {
    const float v = qn[i];
    part += v * v;  // sum over rows of ||q_row||^2 == sum(q^2)
  }
  atomicAdd(&ssum, part);

  float ep0 = 0.0f, ep1 = 0.0f;
  for (int i = threadIdx.x; i < G_ * V_; i += blockDim.x) {
    const float a = avg_raw[i] * (1.0f / (float)(B_ * T_));
    const float term = -a * logf(a + 1e-7f);
    if (i < V_) ep0 += term; else ep1 += term;
  }
  atomicAdd(&ent[0], ep0);
  atomicAdd(&ent[1], ep1);
  __syncthreads();

  if (threadIdx.x == 0) {
    const float loss_l2 = ssum / (float)(B_ * T_ * D_);
    const float loss_d = ((float)V_ - expf(ent[0])) + ((float)V_ - expf(ent[1]));
    out[0] = loss_c[0] + 0.1f * loss_d + 10.0f * loss_l2;
  }
}

// ---------------------------------------------------------------------------
// Workspace layout (floats):
//   Full path  (ws >= ~33.6 MB): S[B*T*T] | qn[B*T] | cn[B*T] | avg[G*V] | loss_c[1]
//   Fallback   (ws >= ~4.3 MB):  S[T*T]   | qn[B*T] | cn[B*T] | avg[G*V] | loss_c[1]
extern "C" void kernel_launch(void* const* d_in, const int* in_sizes, int n_in,
                              void* d_out, int out_size, void* d_ws, size_t ws_size,
                              hipStream_t stream) {
  const float* q = (const float*)d_in[0];           // (B,T,D)
  const float* c = (const float*)d_in[1];           // (B,T,D)
  const float* cb = (const float*)d_in[2];          // (B,T,G,V)
  const int* neg = (const int*)d_in[3];             // (B,T,K)
  const unsigned char* mask = (const unsigned char*)d_in[4];  // (B,T) bool
  float* out = (float*)d_out;

  const size_t tail = (size_t)(2 * B_ * T_ + G_ * V_ + 1);
  const size_t need_full = ((size_t)B_ * T_ * T_ + tail) * sizeof(float);
  const bool full = ws_size >= need_full;
  const size_t s_elems = full ? (size_t)B_ * T_ * T_ : (size_t)T_ * T_;

  float* S = (float*)d_ws;
  float* qn = S + s_elems;
  float* cn = qn + B_ * T_;
  float* avg = cn + B_ * T_;   // G*V floats, loss_c right after
  float* lossc = avg + G_ * V_;

  init_kernel<<<1, 1024, 0, stream>>>(avg);
  norms_kernel<<<(B_ * T_) / 8, 256, 0, stream>>>(q, c, qn, cn);
  diversity_kernel<<<(B_ * T_) / 64, 256, 0, stream>>>(cb, avg);

  if (full) {
    // One GEMM launch over all batches (1024 blocks), then one gather launch.
    gemm_scores<<<dim3(T_ / 64, T_ / 128, B_), 256, 0, stream>>>(
        c, q, S, (long)T_ * D_, (long)T_ * T_);
    contrastive_kernel<<<dim3(T_ / 8, B_), 256, 0, stream>>>(
        S, (long)T_ * T_, neg, mask, qn, cn, lossc, 0);
  } else {
    for (int b = 0; b < B_; ++b) {
      const float* Cb = c + (size_t)b * T_ * D_;
      const float* Qb = q + (size_t)b * T_ * D_;
      gemm_scores<<<dim3(T_ / 64, T_ / 128, 1), 256, 0, stream>>>(Cb, Qb, S, 0, 0);
      contrastive_kernel<<<dim3(T_ / 8, 1), 256, 0, stream>>>(
          S, 0, neg, mask, qn, cn, lossc, b);
    }
  }

  final_kernel<<<1, 256, 0, stream>>>(qn, avg, lossc, out);
}